// SteinSolverAD_52020643889815
// MI455X (gfx1250) — compile-verified
//
#include <hip/hip_runtime.h>
#include <math.h>

// CDNA5 (gfx1250) wave32 types
typedef __attribute__((ext_vector_type(2))) float v2f;
typedef __attribute__((ext_vector_type(8))) float v8f;
typedef __attribute__((ext_vector_type(4))) unsigned int u32x4;
typedef __attribute__((ext_vector_type(8))) int i32x8;
typedef __attribute__((ext_vector_type(4))) int i32x4;

#define N_ITERS 50

#if __has_builtin(__builtin_amdgcn_tensor_load_to_lds) && __has_builtin(__builtin_amdgcn_s_wait_tensorcnt)
#define STEIN_TDM 1
#if __has_include(<hip/amd_detail/amd_gfx1250_TDM.h>)
#define STEIN_TDM_6ARG 1   // therock/clang-23 toolchain: 6-arg builtin
#endif
#endif

// ---------------------------------------------------------------------------
// Zero-fill kernel (Q0 = 0; d_out is poisoned by the harness).
// ---------------------------------------------------------------------------
__global__ __launch_bounds__(256) void stein_zero_f32(float* __restrict__ p, long long n) {
    long long i = (long long)blockIdx.x * 256 + threadIdx.x;
    long long stride = (long long)gridDim.x * 256;
    for (; i < n; i += stride) p[i] = 0.0f;
}

#ifdef STEIN_TDM
// ---------------------------------------------------------------------------
// Issue one TDM 2D tile load: global (tile_h rows x tile_w fp32, row stride
// stride_elems) -> LDS at lds_addr, with hardware row padding so the LDS
// layout is [tile_h][tile_w + pad] (bank-conflict-free for WMMA fragments).
//   pad_icode: pad_interval code (4 = 32 dwords, 6 = 128, 7 = 256)
//   pad_acode: pad_amount  code (3 = 4 dwords,  7 = 8)
// D# packing per CDNA5 ISA ch. 8 (async/tensor).
// ---------------------------------------------------------------------------
__device__ __forceinline__ void tdm_load_tile(
    unsigned lds_addr, const float* gptr,
    unsigned tile_w, unsigned tile_h,
    unsigned long long stride_elems,
    unsigned pad_icode, unsigned pad_acode)
{
    unsigned long long ga = (unsigned long long)(size_t)gptr;
    u32x4 g0;
    g0[0] = 1u;                                   // count=1, user descriptor
    g0[1] = lds_addr;                             // lds_addr (bytes)
    g0[2] = (unsigned)ga;                         // global_addr[31:0]
    g0[3] = (unsigned)((ga >> 32) & 0x01ffffffu)  // global_addr[56:32]
          | (2u << 30);                           // type = 2 ("image")
    const unsigned td0 = 0x7fffffffu;             // huge tensor dims: tiles are
    const unsigned td1 = 0x7fffffffu;             // always in-bounds, no clipping
    i32x8 g1;
    g1[0] = (int)((2u << 16)                      // data_size = 4 bytes
                | (1u << 20)                      // pad_enable
                | (pad_icode << 22)               // pad_interval
                | (pad_acode << 25));             // pad_amount
    g1[1] = (int)((td0 & 0xffffu) << 16);                         // tensor_dim0 lo
    g1[2] = (int)((td0 >> 16) | ((td1 & 0xffffu) << 16));         // dim0 hi | dim1 lo
    g1[3] = (int)((td1 >> 16) | (tile_w << 16));                  // dim1 hi | tile_dim0
    g1[4] = (int)tile_h;                                          // tile_dim1 | tile_dim2=0
    g1[5] = (int)(unsigned)(stride_elems & 0xffffffffu);          // dim0_stride lo
    g1[6] = (int)(unsigned)((stride_elems >> 32) & 0xffffu);      // dim0_stride hi
    g1[7] = 0;
    i32x4 z = {0, 0, 0, 0};
#ifdef STEIN_TDM_6ARG
    i32x8 z8 = {0, 0, 0, 0, 0, 0, 0, 0};
    __builtin_amdgcn_tensor_load_to_lds(g0, g1, z, z, z8, 0);
#else
    __builtin_amdgcn_tensor_load_to_lds(g0, g1, z, z, 0);
#endif
}
#endif // STEIN_TDM

// ---------------------------------------------------------------------------
// Tiled fp32 GEMM on V_WMMA_F32_16X16X4_F32, TDM-staged + double-buffered.
//   D (M x N) = opA(Am) * B [+ Cadd]
//   TRANSA=0: Am is (M x K) row-major, opA = Am   -> LDS A layout [256][36]
//   TRANSA=1: Am is (K x M) row-major, opA = Am^T -> LDS A layout [32][264]
//   B is (K x N) row-major                        -> LDS B layout [32][136]
// Block: 256 threads = 8 waves. Block tile 256x128x32.
// Wave grid 4x2, wave tile 64x64 = 16 WMMA accumulators (16 WMMA / 8 frag loads
// per K=4 step).
// ---------------------------------------------------------------------------
template <bool TRANSA, bool HASC>
__global__ __launch_bounds__(256) void stein_gemm(
    float* __restrict__ D, const float* __restrict__ Am,
    const float* __restrict__ B, const float* __restrict__ Cadd,
    int K, int lda, int ldb, int ldd)
{
    __shared__ __align__(16) float As[2][9216];   // NN: [256][36] | TN: [32][264]
    __shared__ __align__(16) float Bs[2][4352];   // [32][136]

    const int tid   = threadIdx.x;
    const int lane  = tid & 31;
    const int wave  = tid >> 5;
    const int bm    = blockIdx.y * 256;
    const int bn    = blockIdx.x * 128;
    const int wm    = (wave >> 1) * 64;   // 4 wave-rows
    const int wn    = (wave & 1) * 64;    // 2 wave-cols
    const int lhalf = lane >> 4;          // 0: K pair (0,1) | 1: K pair (2,3)
    const int l15   = lane & 15;

    v8f acc[4][4];
#pragma unroll
    for (int im = 0; im < 4; ++im)
#pragma unroll
        for (int in = 0; in < 4; ++in)
            acc[im][in] = (v8f){0.f, 0.f, 0.f, 0.f, 0.f, 0.f, 0.f, 0.f};

    const int nt = K >> 5;                // K / 32 tiles

#ifdef STEIN_TDM
    const unsigned asAddr[2] = { (unsigned)(size_t)&As[0][0], (unsigned)(size_t)&As[1][0] };
    const unsigned bsAddr[2] = { (unsigned)(size_t)&Bs[0][0], (unsigned)(size_t)&Bs[1][0] };

    // Prologue: kick off tile 0 into buffer 0 (2 descriptors: A then B).
    if (wave == 0) {
        if (TRANSA)
            tdm_load_tile(asAddr[0], Am + bm, 256u, 32u,
                          (unsigned long long)lda, 7u, 7u);          // [32][264]
        else
            tdm_load_tile(asAddr[0], Am + (size_t)bm * lda, 32u, 256u,
                          (unsigned long long)lda, 4u, 3u);          // [256][36]
        tdm_load_tile(bsAddr[0], B + bn, 128u, 32u,
                      (unsigned long long)ldb, 6u, 7u);              // [32][136]
    }
#endif

    for (int t = 0; t < nt; ++t) {
        const int cur = t & 1;
        const int kt  = t << 5;

#ifdef STEIN_TDM
        // Issue tile t+1 into the alternate buffer, then wait for tile t.
        if (wave == 0) {
            if (t + 1 < nt) {
                const int ktn = kt + 32;
                if (TRANSA)
                    tdm_load_tile(asAddr[cur ^ 1], Am + (size_t)ktn * lda + bm, 256u, 32u,
                                  (unsigned long long)lda, 7u, 7u);
                else
                    tdm_load_tile(asAddr[cur ^ 1], Am + (size_t)bm * lda + ktn, 32u, 256u,
                                  (unsigned long long)lda, 4u, 3u);
                tdm_load_tile(bsAddr[cur ^ 1], B + (size_t)ktn * ldb + bn, 128u, 32u,
                              (unsigned long long)ldb, 6u, 7u);
                __builtin_amdgcn_s_wait_tensorcnt(2);   // tile t's 2 xfers done (in-order)
            } else {
                __builtin_amdgcn_s_wait_tensorcnt(0);
            }
        }
        __syncthreads();
        const float* __restrict__ Asb = As[cur];
        const float* __restrict__ Bsb = Bs[cur];
#else
        // Fallback: manual staging into buffer 0.
        __syncthreads();
        if (TRANSA) {
            const int kl = tid >> 5, mv = (tid & 31) * 4;   // 32 k-rows x 256 m
#pragma unroll
            for (int p = 0; p < 4; ++p) {
                const int k = kl + p * 8;
                float4 v0 = *(const float4*)(Am + (size_t)(kt + k) * lda + bm + mv);
                float4 v1 = *(const float4*)(Am + (size_t)(kt + k) * lda + bm + mv + 128);
                *(float4*)(&As[0][k * 264 + mv]) = v0;
                *(float4*)(&As[0][k * 264 + mv + 128]) = v1;
            }
        } else {
            const int rl = tid >> 3, kv = (tid & 7) * 4;    // 256 m-rows x 32 k
#pragma unroll
            for (int p = 0; p < 8; ++p) {
                const int m = rl + p * 32;
                float4 v = *(const float4*)(Am + (size_t)(bm + m) * lda + kt + kv);
                *(float4*)(&As[0][m * 36 + kv]) = v;
            }
        }
        {
            const int kl = tid >> 5, nv = (tid & 31) * 4;
#pragma unroll
            for (int p = 0; p < 4; ++p) {
                const int k = kl + p * 8;
                float4 v = *(const float4*)(B + (size_t)(kt + k) * ldb + bn + nv);
                *(float4*)(&Bs[0][k * 136 + nv]) = v;
            }
        }
        __syncthreads();
        const float* __restrict__ Asb = As[0];
        const float* __restrict__ Bsb = Bs[0];
#endif

        // ---- compute: 8 k-steps of K=4, 16 WMMAs each ----
#pragma unroll
        for (int kk = 0; kk < 32; kk += 4) {
            v2f afr[4], bfr[4];
#pragma unroll
            for (int im = 0; im < 4; ++im) {
                if (TRANSA) {
                    const int c = wm + im * 16 + l15;
                    const int r = kk + 2 * lhalf;
                    v2f v; v[0] = Asb[r * 264 + c]; v[1] = Asb[(r + 1) * 264 + c];
                    afr[im] = v;
                } else {
                    const int m = wm + im * 16 + l15;
                    afr[im] = *(const v2f*)(&Asb[m * 36 + kk + 2 * lhalf]);
                }
            }
#pragma unroll
            for (int in = 0; in < 4; ++in) {
                const int c = wn + in * 16 + l15;
                const int r = kk + 2 * lhalf;
                v2f v; v[0] = Bsb[r * 136 + c]; v[1] = Bsb[(r + 1) * 136 + c];
                bfr[in] = v;
            }
#pragma unroll
            for (int im = 0; im < 4; ++im)
#pragma unroll
                for (int in = 0; in < 4; ++in)
                    acc[im][in] = __builtin_amdgcn_wmma_f32_16x16x4_f32(
                        false, afr[im], false, bfr[in],
                        (short)0, acc[im][in], false, false);
        }
        __syncthreads();   // all reads of this buffer done before it is refilled
    }

    // ---- epilogue: C/D layout — VGPR r: lanes 0-15 = (M=r), lanes 16-31 = (M=r+8)
#pragma unroll
    for (int im = 0; im < 4; ++im) {
#pragma unroll
        for (int in = 0; in < 4; ++in) {
            const int col = bn + wn + in * 16 + l15;
#pragma unroll
            for (int r = 0; r < 8; ++r) {
                const int row = bm + wm + im * 16 + r + 8 * lhalf;
                float v = acc[im][in][r];
                if (HASC) v += Cadd[(size_t)row * ldd + col];
                D[(size_t)row * ldd + col] = v;
            }
        }
    }
}

// ---------------------------------------------------------------------------
// Host: RHS = C_F^T C; then 50x { T = A_F^T Q ; Q' = T A + RHS }.
// Ping-pong Q between d_out and workspace so Q_50 lands in d_out.
// ---------------------------------------------------------------------------
extern "C" void kernel_launch(void* const* d_in, const int* in_sizes, int n_in,
                              void* d_out, int out_size, void* d_ws, size_t ws_size,
                              hipStream_t stream) {
    const float* A   = (const float*)d_in[0];  // (n , n )
    const float* A_F = (const float*)d_in[1];  // (nF, nF)
    const float* C   = (const float*)d_in[2];  // (p , n )
    const float* C_F = (const float*)d_in[3];  // (p , nF)

    const int n  = (int)lround(sqrt((double)in_sizes[0]));
    const int nF = (int)lround(sqrt((double)in_sizes[1]));
    const int p  = in_sizes[2] / n;

    float* out  = (float*)d_out;               // Q   (nF x n)
    float* ws   = (float*)d_ws;
    float* RHS  = ws;                          // (nF x n)
    float* T    = RHS + (size_t)nF * n;        // (nF x n)
    float* Qalt = T   + (size_t)nF * n;        // (nF x n)

    const long long total = (long long)nF * n;
    dim3 blk(256);
    dim3 grd((unsigned)(n / 128), (unsigned)(nF / 256));

    // Q0 = 0 in d_out
    stein_zero_f32<<<dim3(2048), blk, 0, stream>>>(out, total);

    // RHS = C_F^T @ C   (A-op: C_F (p x nF) transposed; K = p)
    stein_gemm<true, false><<<grd, blk, 0, stream>>>(RHS, C_F, C, nullptr, p, nF, n, n);

    for (int it = 0; it < N_ITERS; ++it) {
        const float* Qsrc = (it & 1) ? Qalt : out;
        float*       Qdst = (it & 1) ? out  : Qalt;
        // T = A_F^T @ Qsrc
        stein_gemm<true, false><<<grd, blk, 0, stream>>>(T, A_F, Qsrc, nullptr, nF, nF, n, n);
        // Qdst = T @ A + RHS
        stein_gemm<false, true><<<grd, blk, 0, stream>>>(Qdst, T, A, RHS, n, n, n, n);
    }
}